// ShellDescriptor_63702954934357
// MI455X (gfx1250) — compile-verified
//
#include <hip/hip_runtime.h>
#include <math.h>

typedef __attribute__((ext_vector_type(2))) float v2f;
typedef __attribute__((ext_vector_type(8))) float v8f;

#define NN 16
#define NO 128
#define NA 128
#define NPJ 27
#define NF 2
#define NV 30
#define NVP 32                    // v padded to 32 for two WMMA N-tiles
#define M_TOT (NA * NPJ * NF)     // 6912, GEMM M per n
#define K_TOT NO                  // 128, GEMM K
#define OUTC 540                  // f*30*9

// ---------------------------------------------------------------------------
// Stage 1: g[n,o,v] = dense_net(e_j[n,o]); pad v to 32 with zeros.
// ---------------------------------------------------------------------------
__global__ void gj_kernel(const float* __restrict__ e_j,
                          const float* __restrict__ W0, const float* __restrict__ b0,
                          const float* __restrict__ W1, const float* __restrict__ b1,
                          const float* __restrict__ W2, const float* __restrict__ b2,
                          float* __restrict__ g_ws) {
    int idx = blockIdx.x * blockDim.x + threadIdx.x;
    if (idx >= NN * NO) return;
    float x = e_j[idx];
    float h[NV], t[NV];
    for (int j = 0; j < NV; ++j) h[j] = tanhf(x * W0[j] + b0[j]);          // 1->30
    for (int j = 0; j < NV; ++j) {                                          // 30->30 residual
        float s = b1[j];
        for (int i = 0; i < NV; ++i) s += h[i] * W1[i * NV + j];
        t[j] = tanhf(s);
    }
    for (int j = 0; j < NV; ++j) h[j] += t[j];
    for (int j = 0; j < NV; ++j) {                                          // 30->30 residual
        float s = b2[j];
        for (int i = 0; i < NV; ++i) s += h[i] * W2[i * NV + j];
        t[j] = tanhf(s);
    }
    float* g = g_ws + (size_t)idx * NVP;
    for (int j = 0; j < NV; ++j) g[j] = h[j] + t[j];
    g[30] = 0.0f; g[31] = 0.0f;
}

// ---------------------------------------------------------------------------
// Stage 2: u[n, m, v] = (1/128) * sum_k mo_j[n, k, m] * g[n, k, v]
//   m = ((a*27)+p)*2+f in [0,6912); GEMM M=6912, N=32, K=128 per n.
//   One wave per 16-row M-tile; V_WMMA_F32_16X16X4_F32, K chained in steps of 4.
//   A 16x4 layout: lane L holds rows M=L%16, VGPR{0,1} = K = k0+2*(L>>4)+{0,1}.
//   B 4x16 layout: lane L holds col N=L%16,  VGPR{0,1} = K = k0+2*(L>>4)+{0,1}.
//   C/D 16x16:     lane L, VGPR r = element [M = r+8*(L>>4)][N = L%16].
// ---------------------------------------------------------------------------
__global__ void uj_wmma_kernel(const float* __restrict__ mo_j,
                               const float* __restrict__ g_ws,
                               float* __restrict__ u_ws) {
    const int lane   = threadIdx.x & 31;
    const int wave   = threadIdx.x >> 5;
    const int n      = blockIdx.y;
    const int m_tile = blockIdx.x * 8 + wave;        // 432 tiles per n
    const int m0     = m_tile * 16;
    const int mrow   = m0 + (lane & 15);
    const int khalf  = (lane >> 4) * 2;              // 0 or 2
    const int ncol   = lane & 15;

    const float* A = mo_j + (size_t)n * K_TOT * M_TOT;   // [k][m], row stride 6912
    const float* B = g_ws + (size_t)n * K_TOT * NVP;     // [k][v32]

    v8f acc0 = {};   // v = 0..15
    v8f acc1 = {};   // v = 16..31

    for (int k0 = 0; k0 < K_TOT; k0 += 4) {
        const int ka = k0 + khalf;
        v2f a, b0v, b1v;
        a.x   = A[(size_t)ka * M_TOT + mrow];
        a.y   = A[(size_t)(ka + 1) * M_TOT + mrow];
        b0v.x = B[ka * NVP + ncol];
        b0v.y = B[(ka + 1) * NVP + ncol];
        b1v.x = B[ka * NVP + 16 + ncol];
        b1v.y = B[(ka + 1) * NVP + 16 + ncol];
        acc0 = __builtin_amdgcn_wmma_f32_16x16x4_f32(false, a, false, b0v,
                                                     (short)0, acc0, false, false);
        acc1 = __builtin_amdgcn_wmma_f32_16x16x4_f32(false, a, false, b1v,
                                                     (short)0, acc1, false, false);
    }

    const float inv_no = 1.0f / 128.0f;
    float* U = u_ws + ((size_t)n * M_TOT + m0) * NVP;
    for (int r = 0; r < 8; ++r) {
        const int m = r + 8 * (lane >> 4);
        U[(size_t)m * NVP + ncol]      = acc0[r] * inv_no;
        U[(size_t)m * NVP + 16 + ncol] = acc1[r] * inv_no;
    }
}

// ---------------------------------------------------------------------------
// Stage 3: d[n,o,a, (f*30+v)*9+s] = (1/sec_s) * sum_{p in sec_s}
//            mo_i[n,o,a,p,f] * u[n, a*54+p*2+f, v]
//   Block per (a,o,n); u tile (1620 f32) + mo_i tile (54 f32) staged in LDS;
//   540 coalesced stores per block. Pure-bandwidth stage (566 MB output).
// ---------------------------------------------------------------------------
__global__ void d_kernel(const float* __restrict__ mo_i,
                         const float* __restrict__ u_ws,
                         float* __restrict__ out) {
    __shared__ float u_s[NPJ * NF * NV];   // [p][f][v], 1620
    __shared__ float mi_s[NPJ * NF];       // [p][f], 54

    const int a = blockIdx.x, o = blockIdx.y, n = blockIdx.z;
    const int tid = threadIdx.x;

    const float* U = u_ws + ((size_t)(n * NA + a)) * (NPJ * NF) * NVP;
    for (int t = tid; t < NPJ * NF * NV; t += blockDim.x) {
        const int pf = t / NV, v = t % NV;
        u_s[t] = U[pf * NVP + v];
    }
    const float* MI = mo_i + (((size_t)(n * NO + o)) * NA + a) * (NPJ * NF);
    if (tid < NPJ * NF) mi_s[tid] = MI[tid];
    __syncthreads();

    const int off_tab[9] = {0, 1, 2, 3, 6, 9, 12, 17, 22};
    const int sec_tab[9] = {1, 1, 1, 3, 3, 3, 5, 5, 5};

    float* O = out + (((size_t)(n * NO + o)) * NA + a) * OUTC;
    for (int idx = tid; idx < OUTC; idx += blockDim.x) {
        const int s = idx % 9;
        const int v = (idx / 9) % NV;
        const int f = idx / (NV * 9);
        const int off = off_tab[s], sec = sec_tab[s];
        float sum = 0.0f;
        for (int p = off; p < off + sec; ++p)
            sum += mi_s[p * NF + f] * u_s[(p * NF + f) * NV + v];
        O[idx] = sum / (float)sec;
    }
}

// ---------------------------------------------------------------------------
extern "C" void kernel_launch(void* const* d_in, const int* in_sizes, int n_in,
                              void* d_out, int out_size, void* d_ws, size_t ws_size,
                              hipStream_t stream) {
    const float* mo_i = (const float*)d_in[0];
    // d_in[1] = e_i (unused by the reference)
    const float* mo_j = (const float*)d_in[2];
    const float* e_j  = (const float*)d_in[3];
    const float* W0   = (const float*)d_in[4];
    const float* b0   = (const float*)d_in[5];
    const float* W1   = (const float*)d_in[6];
    const float* b1   = (const float*)d_in[7];
    const float* W2   = (const float*)d_in[8];
    const float* b2   = (const float*)d_in[9];
    float* out = (float*)d_out;

    float* g_ws = (float*)d_ws;                        // 16*128*32 f32 = 256 KB
    float* u_ws = g_ws + (size_t)NN * NO * NVP;        // 16*6912*32 f32 = 14.2 MB

    gj_kernel<<<(NN * NO + 255) / 256, 256, 0, stream>>>(e_j, W0, b0, W1, b1, W2, b2, g_ws);

    // 432 M-tiles per n, 8 waves per block -> grid (54, 16)
    uj_wmma_kernel<<<dim3(54, NN), 256, 0, stream>>>(mo_j, g_ws, u_ws);

    d_kernel<<<dim3(NA, NO, NN), 256, 0, stream>>>(mo_i, u_ws, out);
}